// FilterDetections_22308060135971
// MI455X (gfx1250) — compile-verified
//
#include <hip/hip_runtime.h>
#include <stdint.h>

// Problem constants (match reference)
#define BB       8
#define NN       20000
#define CC       20
#define LL       200
#define MAX_DET  300
#define SCORE_THR 0.05f
#define IOU_THR   0.5f
#define NEGV     (-1e9f)
#define NEG_HALF (-5e8f)

#define TPB   256
#define NWAVE (TPB / 32)
#define KPT   80              // 256*80 = 20480 >= 20000 candidates/thread
#define NCAND (CC * MAX_DET)  // 6000
#define K2    24              // 256*24 = 6144 >= 6000

typedef unsigned int u32;
typedef u32 u32x4 __attribute__((ext_vector_type(4)));
typedef int i32x4 __attribute__((ext_vector_type(4)));
typedef int i32x8 __attribute__((ext_vector_type(8)));

#if defined(__HIP_DEVICE_COMPILE__) && __has_builtin(__builtin_amdgcn_tensor_load_to_lds) && __has_builtin(__builtin_amdgcn_s_wait_tensorcnt)
#define HAVE_TDM 1
#else
#define HAVE_TDM 0
#endif

#if HAVE_TDM
// DMA the whole per-batch box table (20000 x 4 f32 = 320,000 B) into LDS with
// one TDM descriptor: 2-D f32 tensor, dim0=8000, dim1=10, stride0=8000,
// tile == whole tensor. Issued by one wave; TDM ignores EXEC.
// clang-23 toolchain signature: (u32x4 g0, i32x8 g1, i32x4 g2, i32x4 g3, i32x8 g4, i32 cpol)
__device__ __forceinline__ void tdm_load_boxes(const float* gsrc, u32 lds_off) {
  uint64_t ga = (uint64_t)(uintptr_t)gsrc;
  u32x4 g0;
  g0.x = 1u;                                            // count=1 (valid D#)
  g0.y = lds_off;                                       // LDS byte address
  g0.z = (u32)ga;                                       // global_addr[31:0]
  g0.w = ((u32)(ga >> 32) & 0x01FFFFFFu) | (2u << 30);  // addr[56:32] | type=2
  const u32 TD0 = 8000u, TD1 = 10u, TIL0 = 8000u, TIL1 = 10u, STR0 = 8000u;
  i32x8 g1;
  g1[0] = (int)(2u << 16);                                          // data_size = 4B
  g1[1] = (int)((TD0 & 0xFFFFu) << 16);                             // tensor_dim0 lo
  g1[2] = (int)(((TD0 >> 16) & 0xFFFFu) | ((TD1 & 0xFFFFu) << 16)); // dim0 hi | dim1 lo
  g1[3] = (int)(((TD1 >> 16) & 0xFFFFu) | ((TIL0 & 0xFFFFu) << 16));// dim1 hi | tile0
  g1[4] = (int)(TIL1 & 0xFFFFu);                                    // tile1 | tile2=0
  g1[5] = (int)STR0;                                                // stride0[31:0]
  g1[6] = 0;                                                        // stride0 hi | stride1 lo
  g1[7] = 0;
  i32x4 z4 = {0, 0, 0, 0};
  i32x8 z8 = {0, 0, 0, 0, 0, 0, 0, 0};
  __builtin_amdgcn_tensor_load_to_lds(g0, g1, z4, z4, z8, 0);
  __builtin_amdgcn_s_wait_tensorcnt(0);
}
#endif

// --------------------------------------------------------------------------
// Kernel 1: per-(batch,class) greedy NMS. grid = B*C blocks, 256 threads.
// Boxes staged once into 320 KB LDS via TDM; per-thread score vector of 80
// f32 kept in VGPRs (fully unrolled); 300 iterations of
// (workgroup argmax -> broadcast winner -> IoU suppression from LDS).
// Writes per-class candidate (score, anchor) lists to workspace.
// --------------------------------------------------------------------------
__global__ __launch_bounds__(TPB, 1) void nms_per_class_kernel(
    const float* __restrict__ boxes,       // (B, N, 4)
    const float* __restrict__ cls,         // (B, N, C)
    float* __restrict__ cand_score,        // (B*C, MAX_DET)
    int* __restrict__ cand_idx) {          // (B*C, MAX_DET)
  __shared__ float shb[NN * 4];            // 320,000 B of LDS (fits 320 KB WGP LDS)
  __shared__ float rv[NWAVE];
  __shared__ int   ri[NWAVE];
  __shared__ float bcast_v;
  __shared__ int   bcast_i;

  const int bc  = blockIdx.x;
  const int b   = bc / CC;
  const int c   = bc % CC;
  const int tid = threadIdx.x;
  const float* gb = boxes + (size_t)b * NN * 4;

#if HAVE_TDM
  if (tid < 32) {
    tdm_load_boxes(gb, (u32)(uintptr_t)(void*)&shb[0]);
  }
#else
  for (int j = tid; j < NN * 4; j += TPB) shb[j] = gb[j];
#endif
  __syncthreads();

  // Load + threshold scores into registers (candidate j = tid + k*256).
  const float* sc = cls + (size_t)b * NN * CC + c;
  __builtin_prefetch(sc, 0, 1);  // global_prefetch_b8
  float work[KPT];
#pragma unroll
  for (int k = 0; k < KPT; ++k) {
    int j = tid + k * TPB;
    float s = NEGV;
    if (j < NN) s = sc[(size_t)j * CC];
    work[k] = (s > SCORE_THR) ? s : NEGV;
  }

  float* csc = cand_score + (size_t)bc * MAX_DET;
  int*   cid = cand_idx   + (size_t)bc * MAX_DET;

  for (int it = 0; it < MAX_DET; ++it) {
    // ---- local argmax over private slots (ascending j => strict '>' keeps lowest idx)
    float bv = -3.4e38f;
    int   bj = 0x7fffffff;
#pragma unroll
    for (int k = 0; k < KPT; ++k) {
      int j = tid + k * TPB;
      float v = work[k];
      if (v > bv) { bv = v; bj = j; }
    }
    // ---- wave32 reduction (tie-break: lowest index, matching jnp.argmax)
    for (int off = 16; off > 0; off >>= 1) {
      float ov = __shfl_xor(bv, off, 32);
      int   oj = __shfl_xor(bj, off, 32);
      if (ov > bv || (ov == bv && oj < bj)) { bv = ov; bj = oj; }
    }
    if ((tid & 31) == 0) { rv[tid >> 5] = bv; ri[tid >> 5] = bj; }
    __syncthreads();
    if (tid == 0) {
      float fv = rv[0];
      int   fj = ri[0];
      for (int w = 1; w < NWAVE; ++w)
        if (rv[w] > fv || (rv[w] == fv && ri[w] < fj)) { fv = rv[w]; fj = ri[w]; }
      bcast_v = fv;
      bcast_i = fj;
      csc[it] = (fv > NEG_HALF) ? fv : NEGV;  // original score when ok, else NEG
      cid[it] = (fj < NN) ? fj : 0;           // always a safe anchor index
    }
    __syncthreads();

    // ---- suppression: IoU of winner vs all private candidates (boxes from LDS)
    int wj = bcast_i;
    if (wj >= NN) wj = 0;
    float wx1 = shb[wj * 4 + 0], wy1 = shb[wj * 4 + 1];
    float wx2 = shb[wj * 4 + 2], wy2 = shb[wj * 4 + 3];
    float wa = (wx2 - wx1) * (wy2 - wy1);
#pragma unroll
    for (int k = 0; k < KPT; ++k) {
      int j  = tid + k * TPB;
      int jj = (j < NN) ? j : 0;
      float x1 = shb[jj * 4 + 0], y1 = shb[jj * 4 + 1];
      float x2 = shb[jj * 4 + 2], y2 = shb[jj * 4 + 3];
      float ix = fminf(wx2, x2) - fmaxf(wx1, x1);
      float iy = fminf(wy2, y2) - fmaxf(wy1, y1);
      float inter = fmaxf(ix, 0.0f) * fmaxf(iy, 0.0f);
      float area  = (x2 - x1) * (y2 - y1);
      float iou   = inter / (wa + area - inter + 1e-8f);
      if (iou > IOU_THR) work[k] = NEGV;   // winner self-suppresses (iou == 1)
    }
    // (no extra barrier needed: next iteration's rv/ri writes come after the
    //  second __syncthreads above; bcast is re-written only after the next first one)
  }
}

// --------------------------------------------------------------------------
// Kernel 2: per-batch top-300 over 6000 candidates + gather. grid = B blocks.
// --------------------------------------------------------------------------
__global__ __launch_bounds__(TPB, 1) void topk_gather_kernel(
    const float* __restrict__ boxes,       // (B, N, 4)
    const float* __restrict__ lcls,        // (B, N, L)
    const float* __restrict__ cand_score,  // (B*C, MAX_DET) == (B, 6000)
    const int* __restrict__ cand_idx,      // (B*C, MAX_DET)
    float* __restrict__ out) {
  __shared__ float rv[NWAVE];
  __shared__ int   ri[NWAVE];
  __shared__ int   bcast_i;
  __shared__ float sel_v[MAX_DET];
  __shared__ int   sel_s[MAX_DET];

  const int b   = blockIdx.x;
  const int tid = threadIdx.x;
  const float* cs_base = cand_score + (size_t)b * NCAND;

  float cs[K2];
#pragma unroll
  for (int k = 0; k < K2; ++k) {
    int s = tid + k * TPB;
    cs[k] = (s < NCAND) ? cs_base[s] : -3.4e38f;  // pads strictly below NEG
  }

  // iterative top-300 (suppress-to -inf => distinct slots; ties -> lowest slot,
  // matching lax.top_k's stable descending order)
  for (int it = 0; it < MAX_DET; ++it) {
    float bv = -3.4e38f;
    int   bs = 0x7fffffff;
#pragma unroll
    for (int k = 0; k < K2; ++k) {
      int s = tid + k * TPB;
      float v = cs[k];
      if (v > bv) { bv = v; bs = s; }
    }
    for (int off = 16; off > 0; off >>= 1) {
      float ov = __shfl_xor(bv, off, 32);
      int   os = __shfl_xor(bs, off, 32);
      if (ov > bv || (ov == bv && os < bs)) { bv = ov; bs = os; }
    }
    if ((tid & 31) == 0) { rv[tid >> 5] = bv; ri[tid >> 5] = bs; }
    __syncthreads();
    if (tid == 0) {
      float fv = rv[0];
      int   fs = ri[0];
      for (int w = 1; w < NWAVE; ++w)
        if (rv[w] > fv || (rv[w] == fv && ri[w] < fs)) { fv = rv[w]; fs = ri[w]; }
      sel_v[it] = fv;
      sel_s[it] = fs;
      bcast_i   = fs;
    }
    __syncthreads();
    int winner = bcast_i;
#pragma unroll
    for (int k = 0; k < K2; ++k)
      if (tid + k * TPB == winner) cs[k] = -3.4e38f;
  }
  __syncthreads();

  // gather: one wave per output row; l_classification argmax over L=200 per row
  float* ob  = out;                               // (B,300,4)
  float* osc = out + (size_t)BB * MAX_DET * 4;    // (B,300)
  int*   olb = (int*)(out + (size_t)BB * MAX_DET * 5);
  float* ols = out + (size_t)BB * MAX_DET * 6;
  int*   oll = (int*)(out + (size_t)BB * MAX_DET * 7);

  const int wid  = tid >> 5;
  const int lane = tid & 31;
  for (int o = wid; o < MAX_DET; o += NWAVE) {
    int   slot  = sel_s[o];
    float sval  = sel_v[o];
    bool  valid = sval > NEG_HALF;
    int cls_id  = slot / MAX_DET;
    int within  = slot - cls_id * MAX_DET;
    int anchor  = cand_idx[((size_t)b * CC + cls_id) * MAX_DET + within];

    const float* lrow = lcls + ((size_t)b * NN + anchor) * LL;
    float mv = -3.4e38f;
    int   ml = 0x7fffffff;
    for (int l = lane; l < LL; l += 32) {
      float v = lrow[l];
      if (v > mv) { mv = v; ml = l; }
    }
    for (int off = 16; off > 0; off >>= 1) {
      float ov = __shfl_xor(mv, off, 32);
      int   ol = __shfl_xor(ml, off, 32);
      if (ov > mv || (ov == mv && ol < ml)) { mv = ov; ml = ol; }
    }

    size_t row = (size_t)b * MAX_DET + o;
    if (lane < 4)
      ob[row * 4 + lane] = valid ? boxes[((size_t)b * NN + anchor) * 4 + lane] : -1.0f;
    if (lane == 0) {
      osc[row] = valid ? sval : -1.0f;
      olb[row] = valid ? cls_id : -1;
      ols[row] = valid ? mv : -1.0f;
      oll[row] = valid ? ml : -1;
    }
  }
}

extern "C" void kernel_launch(void* const* d_in, const int* in_sizes, int n_in,
                              void* d_out, int out_size, void* d_ws, size_t ws_size,
                              hipStream_t stream) {
  (void)in_sizes; (void)n_in; (void)out_size; (void)ws_size;
  const float* boxes = (const float*)d_in[0];  // (B,N,4)
  const float* cls   = (const float*)d_in[1];  // (B,N,C)
  const float* lcls  = (const float*)d_in[2];  // (B,N,L)
  float* out = (float*)d_out;

  float* cand_score = (float*)d_ws;                                   // B*C*300 f32
  int*   cand_idx   = (int*)((char*)d_ws + (size_t)BB * CC * MAX_DET * sizeof(float));

  nms_per_class_kernel<<<BB * CC, TPB, 0, stream>>>(boxes, cls, cand_score, cand_idx);
  topk_gather_kernel<<<BB, TPB, 0, stream>>>(boxes, lcls, cand_score, cand_idx, out);
}